// Net3DSeg_53051436040786
// MI455X (gfx1250) — compile-verified
//
#include <hip/hip_runtime.h>
#include <hip/hip_bf16.h>

typedef float v2f __attribute__((ext_vector_type(2)));
typedef float v8f __attribute__((ext_vector_type(8)));

constexpr int kB  = 4;
constexpr int kP  = 8192;
constexpr int kN  = 32768;   // B*P
constexpr int kC  = 64;
constexpr int kNC = 10;
constexpr int kK  = 12;
constexpr int kH  = 256;
constexpr int kW  = 256;

// ---- output layout (floats): seg_logit | feats_ssp | loss | seg_logit2 ----
constexpr int OUT_SEG1 = 0;
constexpr int OUT_SSP  = kN * kNC;            // 327680
constexpr int OUT_LOSS = OUT_SSP + kN * kC;   // 2424832
constexpr int OUT_SEG2 = OUT_LOSS + 1;

// ---- workspace layout (floats): pre3d | gray | acc[2] ----
constexpr int WS_PRE3D = 0;                   // kN*15
constexpr int WS_GRAY  = kN * 15;
constexpr int WS_ACC   = WS_GRAY + kN;        // 2 floats

// =====================================================================
// Fused GEMM kernel: 4 waves / block, each wave owns a 16-row tile.
//   seg_logit  = feats @ Wl1 + bl1           (N padded 10->16)
//   feats_ssp  = feats @ W3  + b3
//   seg_logit2 = ssp   @ Wl2 + bl2
//   h          = ssp   @ W4  + b4
//   pre_3d     = h     @ W5                  (N padded 15->16)
// All matrix math through v_wmma_f32_16x16x4_f32 (K=64 = 16 steps of 4).
// =====================================================================
__global__ __launch_bounds__(128) void fused_gemm_kernel(
    const float* __restrict__ feats,
    const float* __restrict__ Wl1, const float* __restrict__ bl1,
    const float* __restrict__ Wl2, const float* __restrict__ bl2,
    const float* __restrict__ W3,  const float* __restrict__ b3,
    const float* __restrict__ W4,  const float* __restrict__ b4,
    const float* __restrict__ W5,
    float* __restrict__ out_seg1, float* __restrict__ out_ssp,
    float* __restrict__ out_seg2, float* __restrict__ pre3d)
{
    __shared__ float sW3[kC * kC];
    __shared__ float sW4[kC * kC];
    __shared__ float sW5[kC * 16];
    __shared__ float sL1[kC * 16];
    __shared__ float sL2[kC * 16];
    __shared__ float sb3[kC];
    __shared__ float sb4[kC];
    __shared__ float sbl1[16];
    __shared__ float sbl2[16];
    __shared__ float stage[4 * 16 * kC];   // per-wave 16x64 staging

    const int tid = threadIdx.x;

    for (int i = tid; i < kC * kC; i += 128) { sW3[i] = W3[i]; sW4[i] = W4[i]; }
    for (int i = tid; i < kC * 16; i += 128) {
        int k = i >> 4, n = i & 15;
        sW5[i] = (n < 15)  ? W5[k * 15 + n]   : 0.0f;
        sL1[i] = (n < kNC) ? Wl1[k * kNC + n] : 0.0f;
        sL2[i] = (n < kNC) ? Wl2[k * kNC + n] : 0.0f;
    }
    if (tid < kC) { sb3[tid] = b3[tid]; sb4[tid] = b4[tid]; }
    if (tid < 16) {
        sbl1[tid] = (tid < kNC) ? bl1[tid] : 0.0f;
        sbl2[tid] = (tid < kNC) ? bl2[tid] : 0.0f;
    }
    __syncthreads();

    const int lane = tid & 31;
    const int wave = tid >> 5;
    const int r    = lane & 15;          // N index / A-row within tile
    const int hl   = lane >> 4;          // half-wave selects K pair
    const int row0 = blockIdx.x * 64 + wave * 16;
    float* st = stage + wave * (16 * kC);

    // One 16x16x4 WMMA sweep over K=64 from register A-frags + LDS B.
    auto wmma64 = [&](const float* a0v, const float* a1v,
                      const float* sB, int ldb, int n0, v8f acc) -> v8f {
        #pragma unroll
        for (int kk = 0; kk < 16; ++kk) {
            v2f A; A[0] = a0v[kk]; A[1] = a1v[kk];
            const int kb = 4 * kk + 2 * hl;
            v2f Bv; Bv[0] = sB[kb * ldb + n0 + r];
                    Bv[1] = sB[(kb + 1) * ldb + n0 + r];
            acc = __builtin_amdgcn_wmma_f32_16x16x4_f32(
                false, A, false, Bv, (short)0, acc, false, false);
        }
        return acc;
    };

    // A-fragments of feats (row r, K pairs starting at 4*kk + 2*hl)
    float a0[16], a1[16];
    {
        const float* Ar = feats + (size_t)(row0 + r) * kC + 2 * hl;
        #pragma unroll
        for (int kk = 0; kk < 16; ++kk) { a0[kk] = Ar[4 * kk]; a1[kk] = Ar[4 * kk + 1]; }
    }

    // ---- seg_logit = feats @ Wl1 + bl1 ----
    {
        v8f acc; float bb = sbl1[r];
        #pragma unroll
        for (int i = 0; i < 8; ++i) acc[i] = bb;
        acc = wmma64(a0, a1, sL1, 16, 0, acc);
        if (r < kNC) {
            #pragma unroll
            for (int i = 0; i < 8; ++i)
                out_seg1[(size_t)(row0 + i + 8 * hl) * kNC + r] = acc[i];
        }
    }

    // ---- feats_ssp = feats @ W3 + b3 (4 N-tiles) -> stage + global ----
    #pragma unroll
    for (int nt = 0; nt < 4; ++nt) {
        v8f acc; float bb = sb3[nt * 16 + r];
        #pragma unroll
        for (int i = 0; i < 8; ++i) acc[i] = bb;
        acc = wmma64(a0, a1, sW3, kC, nt * 16, acc);
        #pragma unroll
        for (int i = 0; i < 8; ++i) {
            int m = i + 8 * hl;
            st[m * kC + nt * 16 + r] = acc[i];
            out_ssp[(size_t)(row0 + m) * kC + nt * 16 + r] = acc[i];
        }
    }

    // A-fragments of feats_ssp (same-wave LDS ops are in-order: no barrier)
    float c0[16], c1[16];
    #pragma unroll
    for (int kk = 0; kk < 16; ++kk) {
        c0[kk] = st[r * kC + 4 * kk + 2 * hl];
        c1[kk] = st[r * kC + 4 * kk + 2 * hl + 1];
    }

    // ---- seg_logit2 = ssp @ Wl2 + bl2 ----
    {
        v8f acc; float bb = sbl2[r];
        #pragma unroll
        for (int i = 0; i < 8; ++i) acc[i] = bb;
        acc = wmma64(c0, c1, sL2, 16, 0, acc);
        if (r < kNC) {
            #pragma unroll
            for (int i = 0; i < 8; ++i)
                out_seg2[(size_t)(row0 + i + 8 * hl) * kNC + r] = acc[i];
        }
    }

    // ---- h = ssp @ W4 + b4 -> stage (overwrite) ----
    #pragma unroll
    for (int nt = 0; nt < 4; ++nt) {
        v8f acc; float bb = sb4[nt * 16 + r];
        #pragma unroll
        for (int i = 0; i < 8; ++i) acc[i] = bb;
        acc = wmma64(c0, c1, sW4, kC, nt * 16, acc);
        #pragma unroll
        for (int i = 0; i < 8; ++i)
            st[(i + 8 * hl) * kC + nt * 16 + r] = acc[i];
    }

    // A-fragments of h
    float d0[16], d1[16];
    #pragma unroll
    for (int kk = 0; kk < 16; ++kk) {
        d0[kk] = st[r * kC + 4 * kk + 2 * hl];
        d1[kk] = st[r * kC + 4 * kk + 2 * hl + 1];
    }

    // ---- pre_3d = h @ W5 (no bias) ----
    {
        v8f acc;
        #pragma unroll
        for (int i = 0; i < 8; ++i) acc[i] = 0.0f;
        acc = wmma64(d0, d1, sW5, 16, 0, acc);
        if (r < 15) {
            #pragma unroll
            for (int i = 0; i < 8; ++i)
                pre3d[(size_t)(row0 + i + 8 * hl) * 15 + r] = acc[i];
        }
    }
}

// =====================================================================
// Gather rgb from img, compute gray, accumulate mean|rgb_pre - rgb|.
// =====================================================================
__global__ __launch_bounds__(256) void gather_gray_kernel(
    const float* __restrict__ img, const int* __restrict__ ind,
    const float* __restrict__ pre3d, float* __restrict__ gray,
    float* __restrict__ acc)
{
    int p = blockIdx.x * 256 + threadIdx.x;
    if (p >= kN) return;
    int b  = p >> 13;                 // p / kP
    int i0 = ind[2 * p];
    int i1 = ind[2 * p + 1];
    const float* im = img + (size_t)b * 3 * kH * kW;
    float rr = im[(size_t)i0 * kW + i1];
    float gg = im[(size_t)kH * kW + (size_t)i0 * kW + i1];
    float bb = im[(size_t)2 * kH * kW + (size_t)i0 * kW + i1];
    gray[p] = 0.299f * rr + 0.587f * gg + 0.114f * bb;
    const float* pr = pre3d + (size_t)p * 15;
    float s = fabsf(pr[0] - rr) + fabsf(pr[1] - gg) + fabsf(pr[2] - bb);
    atomicAdd(acc + 0, s);
}

// =====================================================================
// Brute-force KNN (per-batch, K=12, ascending distance, lowest-index
// tie-break via strict <) + grad L1 accumulation.
// =====================================================================
__global__ __launch_bounds__(256) void knn_grad_kernel(
    const float* __restrict__ xyz, const float* __restrict__ gray,
    const float* __restrict__ pre3d, float* __restrict__ acc)
{
    __shared__ float4 tile[256];
    const int b  = blockIdx.x >> 5;                  // kP/256 = 32 blocks/batch
    const int q  = ((blockIdx.x & 31) << 8) + threadIdx.x;
    const int gq = b * kP + q;

    const float qx = xyz[3 * gq], qy = xyz[3 * gq + 1], qz = xyz[3 * gq + 2];
    const float sqq = qx * qx + qy * qy + qz * qz;

    float bd[kK]; int bi[kK];
    #pragma unroll
    for (int k = 0; k < kK; ++k) { bd[k] = 3.4e38f; bi[k] = 0; }

    for (int t = 0; t < kP / 256; ++t) {
        __syncthreads();
        {
            int gj = b * kP + (t << 8) + threadIdx.x;
            float x = xyz[3 * gj], y = xyz[3 * gj + 1], z = xyz[3 * gj + 2];
            tile[threadIdx.x] = make_float4(x, y, z, x * x + y * y + z * z);
        }
        __syncthreads();
        for (int jj = 0; jj < 256; ++jj) {
            float4 c = tile[jj];
            float dot = qx * c.x + qy * c.y + qz * c.z;
            float d = sqq - 2.0f * dot + c.w;      // same formula as reference
            if (d < bd[kK - 1]) {
                bd[kK - 1] = d;
                bi[kK - 1] = (t << 8) + jj;
                #pragma unroll
                for (int k = kK - 1; k > 0; --k) {
                    if (bd[k] < bd[k - 1]) {
                        float td = bd[k]; bd[k] = bd[k - 1]; bd[k - 1] = td;
                        int   ti = bi[k]; bi[k] = bi[k - 1]; bi[k - 1] = ti;
                    }
                }
            }
        }
    }

    const float g = gray[gq];
    float s = 0.0f;
    #pragma unroll
    for (int k = 0; k < kK; ++k) {
        float gn = gray[b * kP + bi[k]];
        s += fabsf((gn - g) - pre3d[(size_t)gq * 15 + 3 + k]);
    }
    atomicAdd(acc + 1, s);
}

__global__ void zero_acc_kernel(float* __restrict__ acc) {
    if (blockIdx.x == 0 && threadIdx.x < 2) acc[threadIdx.x] = 0.0f;
}

__global__ void finalize_kernel(const float* __restrict__ acc,
                                float* __restrict__ out_loss) {
    if (blockIdx.x == 0 && threadIdx.x == 0) {
        float disstance = acc[0] * (1.0f / (float)(kN * 3));
        float grad_dis  = acc[1] * (1.0f / (float)(kN * kK));
        out_loss[0] = disstance + 0.1f * grad_dis;
    }
}

// =====================================================================
extern "C" void kernel_launch(void* const* d_in, const int* in_sizes, int n_in,
                              void* d_out, int out_size, void* d_ws, size_t ws_size,
                              hipStream_t stream) {
    const float* feats = (const float*)d_in[0];
    const float* xyz   = (const float*)d_in[1];
    const float* img   = (const float*)d_in[2];
    const int*   ind   = (const int*)  d_in[3];
    const float* Wl1   = (const float*)d_in[4];
    const float* bl1   = (const float*)d_in[5];
    const float* Wl2   = (const float*)d_in[6];
    const float* bl2   = (const float*)d_in[7];
    const float* W3    = (const float*)d_in[8];
    const float* b3    = (const float*)d_in[9];
    const float* W4    = (const float*)d_in[10];
    const float* b4    = (const float*)d_in[11];
    const float* W5    = (const float*)d_in[12];

    float* out      = (float*)d_out;
    float* out_seg1 = out + OUT_SEG1;
    float* out_ssp  = out + OUT_SSP;
    float* out_loss = out + OUT_LOSS;
    float* out_seg2 = out + OUT_SEG2;

    float* ws    = (float*)d_ws;
    float* pre3d = ws + WS_PRE3D;
    float* gray  = ws + WS_GRAY;
    float* acc   = ws + WS_ACC;

    zero_acc_kernel<<<1, 32, 0, stream>>>(acc);
    fused_gemm_kernel<<<kN / 64, 128, 0, stream>>>(
        feats, Wl1, bl1, Wl2, bl2, W3, b3, W4, b4, W5,
        out_seg1, out_ssp, out_seg2, pre3d);
    gather_gray_kernel<<<kN / 256, 256, 0, stream>>>(img, ind, pre3d, gray, acc);
    knn_grad_kernel<<<kN / 256, 256, 0, stream>>>(xyz, gray, pre3d, acc);
    finalize_kernel<<<1, 32, 0, stream>>>(acc, out_loss);
}